// GraphFusion_85710367359259
// MI455X (gfx1250) — compile-verified
//
#include <hip/hip_runtime.h>
#include <hip/hip_bf16.h>
#include <math.h>

// ---------------- problem constants (from reference) ----------------
#define N_NODES 20000
#define N_EDGES 640000
#define E_TOT   (N_EDGES + N_NODES)   // with self loops = 660000
#define F_IN    256
#define HID     64
#define H1      4                      // heads conv1
#define C1      (H1 * HID)             // 256
#define N_GRAPHS 32
#define NEG_SLOPE 0.2f

typedef float v2f __attribute__((ext_vector_type(2)));
typedef float v8f __attribute__((ext_vector_type(8)));

// ---------------- helpers ----------------
__device__ __forceinline__ int   f2ord(float f) { int b = __float_as_int(f); return b >= 0 ? b : (b ^ 0x7FFFFFFF); }
__device__ __forceinline__ float ord2f(int k)   { return __int_as_float(k >= 0 ? k : (k ^ 0x7FFFFFFF)); }
__device__ __forceinline__ float lrelu(float x) { return x > 0.f ? x : NEG_SLOPE * x; }

// ---------------- init kernels ----------------
__global__ void fill_f32(float* p, float v, int n) {
    int i = blockIdx.x * blockDim.x + threadIdx.x;
    if (i < n) p[i] = v;
}
__global__ void fill_i32(int* p, int v, int n) {
    int i = blockIdx.x * blockDim.x + threadIdx.x;
    if (i < n) p[i] = v;
}

// ---------------- fp32 WMMA GEMM: C[M,N] = X[M,K] @ W[N,K]^T ----------------
// One wave per 16x16 tile, blockDim = 32 (wave32). M % 16 == 0, N % 16 == 0, K % 4 == 0.
// V_WMMA_F32_16X16X4_F32 lane layouts per CDNA5 ISA §7.12.2:
//   A 16x4 : lanes 0-15 -> M=lane, K={0,1} in v0/v1 ; lanes 16-31 -> M=lane-16, K={2,3}
//   B 4x16 : lanes 0-15 -> N=lane, K={0,1}          ; lanes 16-31 -> N=lane-16, K={2,3}
//   C 16x16: vgpr r -> M = r (lanes 0-15) / r+8 (lanes 16-31), N = lane&15
__global__ void wmma_gemm_xWt(const float* __restrict__ X, const float* __restrict__ W,
                              float* __restrict__ C, int M, int K, int N) {
    const int ntN  = N >> 4;
    const int tile = blockIdx.x;
    const int mt   = tile / ntN;
    const int nt   = tile % ntN;
    const int lane = threadIdx.x;
    const int half = lane >> 4;      // 0: K-offset 0, 1: K-offset 2
    const int l    = lane & 15;

    const float* arow = X + (size_t)(mt * 16 + l) * K + half * 2;
    const float* brow = W + (size_t)(nt * 16 + l) * K + half * 2;

    v8f acc = {0.f, 0.f, 0.f, 0.f, 0.f, 0.f, 0.f, 0.f};
    for (int k = 0; k < K; k += 4) {
        v2f a = *(const v2f*)(arow + k);
        v2f b = *(const v2f*)(brow + k);
        acc = __builtin_amdgcn_wmma_f32_16x16x4_f32(
            /*neg_a=*/false, a, /*neg_b=*/false, b,
            /*c_mod=*/(short)0, acc, /*reuse_a=*/false, /*reuse_b=*/false);
    }

    float* crow = C + (size_t)(mt * 16 + half * 8) * N + nt * 16 + l;
#pragma unroll
    for (int r = 0; r < 8; ++r) crow[(size_t)r * N] = acc[r];
}

// ---------------- per-node attention dot products ----------------
// as_out[n*heads+h] = sum_c h[n, h*64+c] * att_s[h*64+c]   (and same for att_d)
__global__ void attn_dots(const float* __restrict__ h, const float* __restrict__ att_s,
                          const float* __restrict__ att_d, float* __restrict__ as_out,
                          float* __restrict__ ad_out, int n, int heads) {
    int i = blockIdx.x * blockDim.x + threadIdx.x;
    if (i >= n * heads) return;
    int node = i / heads, hd = i % heads;
    const float* hv = h + (size_t)node * heads * HID + hd * HID;
    const float* sv = att_s + hd * HID;
    const float* dv = att_d + hd * HID;
    float s = 0.f, d = 0.f;
#pragma unroll 4
    for (int c = 0; c < HID; ++c) { float v = hv[c]; s += v * sv[c]; d += v * dv[c]; }
    as_out[i] = s;
    ad_out[i] = d;
}

// ---------------- segment-softmax pass 1: atomic max of logits per dst ----------------
__global__ void edge_max(const int* __restrict__ esrc, const int* __restrict__ edst,
                         const float* __restrict__ as_, const float* __restrict__ ad_,
                         int* __restrict__ m, int heads) {
    int i = blockIdx.x * blockDim.x + threadIdx.x;
    if (i >= E_TOT * heads) return;
    int e = i / heads, hd = i % heads;
    int s, d;
    if (e < N_EDGES) { s = esrc[e]; d = edst[e]; } else { s = d = e - N_EDGES; }
    float lg = lrelu(as_[s * heads + hd] + ad_[d * heads + hd]);
    atomicMax(&m[d * heads + hd], f2ord(lg));
}

// ---------------- segment-softmax pass 2: ex = exp(e - m[dst]); atomic sum ----------------
__global__ void edge_exp(const int* __restrict__ esrc, const int* __restrict__ edst,
                         const float* __restrict__ as_, const float* __restrict__ ad_,
                         const int* __restrict__ m, float* __restrict__ exb,
                         float* __restrict__ sb, int heads) {
    int i = blockIdx.x * blockDim.x + threadIdx.x;
    if (i >= E_TOT * heads) return;
    int e = i / heads, hd = i % heads;
    int s, d;
    if (e < N_EDGES) { s = esrc[e]; d = edst[e]; } else { s = d = e - N_EDGES; }
    float lg = lrelu(as_[s * heads + hd] + ad_[d * heads + hd]);
    float ex = expf(lg - ord2f(m[d * heads + hd]));
    exb[i] = ex;
    atomicAdd(&sb[d * heads + hd], ex);
}

// ---------------- weighted scatter-aggregate: out[dst] += alpha * h[src] ----------------
// one wave per edge; lane strides over channels (coalesced), f32 atomics land in L2
__global__ void edge_aggregate(const int* __restrict__ esrc, const int* __restrict__ edst,
                               const float* __restrict__ exb, const float* __restrict__ sb,
                               const float* __restrict__ h, float* __restrict__ out,
                               int heads, int ch_total) {
    int wave = (blockIdx.x * blockDim.x + threadIdx.x) >> 5;
    int lane = threadIdx.x & 31;
    if (wave >= E_TOT) return;
    int sn, dn;
    if (wave < N_EDGES) { sn = esrc[wave]; dn = edst[wave]; } else { sn = dn = wave - N_EDGES; }
    const float* hs = h + (size_t)sn * ch_total;
    float*       od = out + (size_t)dn * ch_total;
    for (int c = lane; c < ch_total; c += 32) {
        int hd = c >> 6;  // 64 channels per head
        float alpha = exb[(size_t)wave * heads + hd] / (sb[dn * heads + hd] + 1e-16f);
        atomicAdd(&od[c], alpha * hs[c]);
    }
}

// ---------------- bias + ELU (in place ok) ----------------
__global__ void bias_elu(float* __restrict__ x, const float* __restrict__ bias, int n, int ch) {
    int i = blockIdx.x * blockDim.x + threadIdx.x;
    if (i >= n * ch) return;
    float v = x[i] + bias[i % ch];
    x[i] = v > 0.f ? v : (expf(v) - 1.f);
}

// ---------------- global mean pool (accumulate) ----------------
__global__ void pool_accum(const float* __restrict__ h, const int* __restrict__ batch,
                           float* __restrict__ sums, float* __restrict__ cnt) {
    int i = blockIdx.x * blockDim.x + threadIdx.x;
    if (i >= N_NODES * HID) return;
    int node = i >> 6, c = i & 63;
    int g = batch[node];
    atomicAdd(&sums[g * HID + c], h[i]);
    if (c == 0) atomicAdd(&cnt[g], 1.f);
}

// ---------------- classifier: Linear(64,32)+ReLU -> Linear(32,1) -> Sigmoid ----------------
__global__ void classify(const float* __restrict__ sums, const float* __restrict__ cnt,
                         const float* __restrict__ Wc1, const float* __restrict__ bc1,
                         const float* __restrict__ Wc2, const float* __restrict__ bc2,
                         float* __restrict__ out) {
    int g = threadIdx.x;
    if (g >= N_GRAPHS) return;
    float inv = 1.f / fmaxf(cnt[g], 1.f);
    float gm[HID];
#pragma unroll
    for (int c = 0; c < HID; ++c) gm[c] = sums[g * HID + c] * inv;
    float acc2 = bc2[0];
    for (int j = 0; j < 32; ++j) {
        float z = bc1[j];
#pragma unroll 8
        for (int c = 0; c < HID; ++c) z += gm[c] * Wc1[j * HID + c];
        acc2 += fmaxf(z, 0.f) * Wc2[j];
    }
    out[g] = 1.f / (1.f + expf(-acc2));
}

// ---------------- orchestration ----------------
extern "C" void kernel_launch(void* const* d_in, const int* in_sizes, int n_in,
                              void* d_out, int out_size, void* d_ws, size_t ws_size,
                              hipStream_t stream) {
    (void)in_sizes; (void)n_in; (void)out_size; (void)ws_size;
    const float* x    = (const float*)d_in[0];
    const int*   ei   = (const int*)d_in[1];     // [2, E]: src then dst
    const int*   batch= (const int*)d_in[2];
    const float* W1   = (const float*)d_in[3];
    const float* as1  = (const float*)d_in[4];
    const float* ad1  = (const float*)d_in[5];
    const float* b1   = (const float*)d_in[6];
    const float* W2   = (const float*)d_in[7];
    const float* as2  = (const float*)d_in[8];
    const float* ad2  = (const float*)d_in[9];
    const float* b2   = (const float*)d_in[10];
    const float* Wc1  = (const float*)d_in[11];
    const float* bc1  = (const float*)d_in[12];
    const float* Wc2  = (const float*)d_in[13];
    const float* bc2  = (const float*)d_in[14];
    const int* esrc = ei;
    const int* edst = ei + N_EDGES;

    // workspace layout (floats)
    float* ws = (float*)d_ws;
    size_t off = 0;
    float* h1   = ws + off; off += (size_t)N_NODES * C1;       // conv1 features
    float* asv1 = ws + off; off += (size_t)N_NODES * H1;
    float* adv1 = ws + off; off += (size_t)N_NODES * H1;
    int*   m1   = (int*)(ws + off); off += (size_t)N_NODES * H1;
    float* s1   = ws + off; off += (size_t)N_NODES * H1;
    float* ex1  = ws + off; off += (size_t)E_TOT * H1;
    float* o1   = ws + off; off += (size_t)N_NODES * C1;       // conv1 output
    float* h2   = ws + off; off += (size_t)N_NODES * HID;
    float* asv2 = ws + off; off += (size_t)N_NODES;
    float* adv2 = ws + off; off += (size_t)N_NODES;
    int*   m2   = (int*)(ws + off); off += (size_t)N_NODES;
    float* s2   = ws + off; off += (size_t)N_NODES;
    float* ex2  = ws + off; off += (size_t)E_TOT;
    float* o2   = ws + off; off += (size_t)N_NODES * HID;      // conv2 output
    float* gsum = ws + off; off += (size_t)N_GRAPHS * HID;
    float* gcnt = ws + off; off += (size_t)N_GRAPHS;

    const int TB = 256;
    const int INT_NEG = (int)0x80000000;
    auto cdiv = [](long long a, long long b) { return (int)((a + b - 1) / b); };

    // ---------- conv1 ----------
    fill_f32<<<cdiv((size_t)N_NODES * C1, TB), TB, 0, stream>>>(o1, 0.f, N_NODES * C1);
    fill_f32<<<cdiv(N_NODES * H1, TB), TB, 0, stream>>>(s1, 0.f, N_NODES * H1);
    fill_i32<<<cdiv(N_NODES * H1, TB), TB, 0, stream>>>(m1, INT_NEG, N_NODES * H1);

    // h1 = x @ W1^T   (1250 x 16 tiles, one wave each)
    wmma_gemm_xWt<<<(N_NODES / 16) * (C1 / 16), 32, 0, stream>>>(x, W1, h1, N_NODES, F_IN, C1);

    attn_dots<<<cdiv(N_NODES * H1, TB), TB, 0, stream>>>(h1, as1, ad1, asv1, adv1, N_NODES, H1);
    edge_max<<<cdiv((size_t)E_TOT * H1, TB), TB, 0, stream>>>(esrc, edst, asv1, adv1, m1, H1);
    edge_exp<<<cdiv((size_t)E_TOT * H1, TB), TB, 0, stream>>>(esrc, edst, asv1, adv1, m1, ex1, s1, H1);
    edge_aggregate<<<cdiv((size_t)E_TOT * 32, TB), TB, 0, stream>>>(esrc, edst, ex1, s1, h1, o1, H1, C1);
    bias_elu<<<cdiv((size_t)N_NODES * C1, TB), TB, 0, stream>>>(o1, b1, N_NODES, C1);

    // ---------- conv2 ----------
    fill_f32<<<cdiv((size_t)N_NODES * HID, TB), TB, 0, stream>>>(o2, 0.f, N_NODES * HID);
    fill_f32<<<cdiv(N_NODES, TB), TB, 0, stream>>>(s2, 0.f, N_NODES);
    fill_i32<<<cdiv(N_NODES, TB), TB, 0, stream>>>(m2, INT_NEG, N_NODES);

    // h2 = o1 @ W2^T   (1250 x 4 tiles)
    wmma_gemm_xWt<<<(N_NODES / 16) * (HID / 16), 32, 0, stream>>>(o1, W2, h2, N_NODES, C1, HID);

    attn_dots<<<cdiv(N_NODES, TB), TB, 0, stream>>>(h2, as2, ad2, asv2, adv2, N_NODES, 1);
    edge_max<<<cdiv(E_TOT, TB), TB, 0, stream>>>(esrc, edst, asv2, adv2, m2, 1);
    edge_exp<<<cdiv(E_TOT, TB), TB, 0, stream>>>(esrc, edst, asv2, adv2, m2, ex2, s2, 1);
    edge_aggregate<<<cdiv((size_t)E_TOT * 32, TB), TB, 0, stream>>>(esrc, edst, ex2, s2, h2, o2, 1, HID);
    bias_elu<<<cdiv((size_t)N_NODES * HID, TB), TB, 0, stream>>>(o2, b2, N_NODES, HID);

    // ---------- pool + classifier ----------
    fill_f32<<<cdiv(N_GRAPHS * HID, TB), TB, 0, stream>>>(gsum, 0.f, N_GRAPHS * HID);
    fill_f32<<<1, TB, 0, stream>>>(gcnt, 0.f, N_GRAPHS);
    pool_accum<<<cdiv((size_t)N_NODES * HID, TB), TB, 0, stream>>>(o2, batch, gsum, gcnt);
    classify<<<1, 32, 0, stream>>>(gsum, gcnt, Wc1, bc1, Wc2, bc2, (float*)d_out);
}